// VinaModel_12223476925082
// MI455X (gfx1250) — compile-verified
//
#include <hip/hip_runtime.h>
#include <math.h>

// Problem constants (match the reference)
#define NATOMS 4096
#define NCELLS 8000
#define KPC    5     // atoms gathered per cell
#define MNB    8     // neighbors kept per atom
#define NBRC   26    // neighbor cells per cell

typedef float v2f __attribute__((ext_vector_type(2)));
typedef float v8f __attribute__((ext_vector_type(8)));

// cells[n] = (g[b], g[c], g[a]) with n = a*400 + b*20 + c, g[i] = i - 10
// (derived from transpose(meshgrid(g,g,g,'xy')).reshape(-1,3))

// Branchless conditional swap for a descending, stable top-k chain.
// Strict '>' + ascending-index scan order reproduces JAX top_k tie order.
// Lowers to v_cndmask pairs (no EXEC divergence).
__device__ __forceinline__ void csel_swap(float& v, int& j, float& sv, int& sj) {
  const bool  gt = v > sv;
  const float nv = gt ? sv : v;
  const int   nj = gt ? sj : j;
  sv = gt ? v : sv;
  sj = gt ? j : sj;
  v = nv; j = nj;
}

// ---------------------------------------------------------------------------
// Kernel A: for each cell, top-5 FARTHEST atoms (faithful to reference top_k).
// V_WMMA_F32_16X16X4_F32 computes the -2 * cells . coords^T cross term.
// Block = 128 threads (4 waves); each wave covers a disjoint 1024-atom range
// for the block's 16-cell tile; per-wave top-5 lists merged stably at the end.
// ---------------------------------------------------------------------------
__global__ __launch_bounds__(128) void topk_atoms_per_cell_kernel(
    const float* __restrict__ coords, int* __restrict__ atoms_in_cells) {
  const int tid  = threadIdx.x;
  const int wave = tid >> 5;
  const int lane = tid & 31;
  const int lo   = lane & 15;
  const int hi   = lane >> 4;
  const int cellBase = blockIdx.x * 16;

  __shared__ float s_cx[16], s_cy[16], s_cz[16], s_cn[16];
  __shared__ float s_dist[4][16][16];
  __shared__ float s_candV[16][4 * KPC];
  __shared__ int   s_candI[16][4 * KPC];

  if (tid < 16) {
    const int n = cellBase + tid;
    const int a = n / 400;
    const int r = n - a * 400;
    const int b = r / 20;
    const int c = r - b * 20;
    const float cx = (float)(b - 10);
    const float cy = (float)(c - 10);
    const float cz = (float)(a - 10);
    s_cx[tid] = cx; s_cy[tid] = cy; s_cz[tid] = cz;
    s_cn[tid] = cx * cx + cy * cy + cz * cz;
  }
  __syncthreads();

  // A-matrix (16x4 f32, K padded to 4 with zero):
  // lanes 0-15 hold (K=0,K=1) = (x,y); lanes 16-31 hold (K=2,K=3) = (z,0).
  v2f amat;
  amat.x = hi ? s_cz[lo] : s_cx[lo];
  amat.y = hi ? 0.0f     : s_cy[lo];

  float topv[KPC]; int topi[KPC];
#pragma unroll
  for (int s = 0; s < KPC; ++s) { topv[s] = -__builtin_inff(); topi[s] = 0; }

  const int atom0 = wave * (NATOMS / 4);
#pragma unroll 2
  for (int ch = 0; ch < (NATOMS / 4) / 16; ++ch) {
    const int aBase = atom0 + ch * 16;
    const int ai = aBase + lo;
    const float ax = coords[3 * ai + 0];
    const float ay = coords[3 * ai + 1];
    const float az = coords[3 * ai + 2];
    const float an = ax * ax + ay * ay + az * az;

    // B-matrix (4x16 f32), mirrored half-wave layout.
    v2f bmat;
    bmat.x = hi ? az : ax;
    bmat.y = hi ? 0.0f : ay;

    v8f acc = {};
    acc = __builtin_amdgcn_wmma_f32_16x16x4_f32(
        /*neg_a=*/false, amat, /*neg_b=*/false, bmat,
        /*c_mod=*/(short)0, acc, /*reuse_a=*/false, /*reuse_b=*/false);

    // C layout: element (M = v + 8*hi, N = lo). dist = |cell|^2 + |atom|^2 - 2*dot
#pragma unroll
    for (int v = 0; v < 8; ++v) {
      const int row = v + 8 * hi;
      s_dist[wave][row][lo] = s_cn[row] + an - 2.0f * acc[v];
    }

    // Same-wave LDS write->read is in-order (DS pipeline); lanes 0-15 each
    // scan their cell row with a branchless stable top-5 chain.
    if (hi == 0) {
#pragma unroll
      for (int ncol = 0; ncol < 16; ++ncol) {
        float v = s_dist[wave][lo][ncol];
        int   j = aBase + ncol;
#pragma unroll
        for (int s = 0; s < KPC; ++s) csel_swap(v, j, topv[s], topi[s]);
      }
    }
  }

  if (hi == 0) {
#pragma unroll
    for (int s = 0; s < KPC; ++s) {
      s_candV[lo][wave * KPC + s] = topv[s];
      s_candI[lo][wave * KPC + s] = topi[s];
    }
  }
  __syncthreads();

  // Stable merge: waves are in ascending atom-index order, lists are
  // descending-value / ascending-index, so strict '>' keeps JAX tie order.
  if (tid < 16) {
    float bv[KPC]; int bi[KPC];
#pragma unroll
    for (int s = 0; s < KPC; ++s) { bv[s] = -__builtin_inff(); bi[s] = 0; }
    for (int k = 0; k < 4 * KPC; ++k) {
      float v = s_candV[tid][k];
      int   j = s_candI[tid][k];
#pragma unroll
      for (int s = 0; s < KPC; ++s) csel_swap(v, j, bv[s], bi[s]);
    }
#pragma unroll
    for (int s = 0; s < KPC; ++s)
      atoms_in_cells[(cellBase + tid) * KPC + s] = bi[s];
  }
}

// ---------------------------------------------------------------------------
// Kernel B: closest lattice cell per atom (analytic argmin, tie -> lower idx).
// ---------------------------------------------------------------------------
__device__ __forceinline__ int nearest_grid_idx(float t) {
  const float u = t + 10.0f;
  int i0 = (int)floorf(u);
  i0 = i0 < 0 ? 0 : (i0 > 19 ? 19 : i0);
  const int i1 = i0 + 1 > 19 ? 19 : i0 + 1;
  float d0 = u - (float)i0; d0 *= d0;
  float d1 = (float)i1 - u; d1 *= d1;
  return (d0 <= d1) ? i0 : i1;   // tie -> lower index, as argmin does
}

__global__ __launch_bounds__(128) void cells_for_atoms_kernel(
    const float* __restrict__ coords, int* __restrict__ cfa) {
  const int i = blockIdx.x * blockDim.x + threadIdx.x;
  if (i >= NATOMS) return;
  const int bx = nearest_grid_idx(coords[3 * i + 0]);  // x -> b index
  const int by = nearest_grid_idx(coords[3 * i + 1]);  // y -> c index
  const int bz = nearest_grid_idx(coords[3 * i + 2]);  // z -> a index
  cfa[i] = bz * 400 + bx * 20 + by;
}

// ---------------------------------------------------------------------------
// Kernel C: per cell, top-26 FARTHEST cells. Lattice distances are exact
// small integers in f32, so integer math reproduces the reference ties.
// Register-resident 26-deep stable chain behind a threshold early-out
// (expected insertions ~ 26*ln(8000) ~ 230 per thread).
// ---------------------------------------------------------------------------
__global__ __launch_bounds__(128) void nbr_cells_kernel(int* __restrict__ nbr_cells) {
  const int c = blockIdx.x * blockDim.x + threadIdx.x;
  if (c >= NCELLS) return;
  const int a0 = c / 400;
  const int r0 = c - a0 * 400;
  const int b0 = r0 / 20;
  const int c0 = r0 - b0 * 20;

  float tv[NBRC]; int ti[NBRC];
#pragma unroll
  for (int s = 0; s < NBRC; ++s) { tv[s] = -1.0f; ti[s] = 0; }
  float thresh = -1.0f;

  int m = 0;
  for (int am = 0; am < 20; ++am) {
    const int dz = am - a0;
    const int dz2 = dz * dz;
    for (int bm = 0; bm < 20; ++bm) {
      const int dx = bm - b0;
      const int dxz2 = dz2 + dx * dx;
      for (int cm = 0; cm < 20; ++cm, ++m) {
        const int dy = cm - c0;
        const float d = (float)(dxz2 + dy * dy);
        if (d > thresh) {                 // strict: equal-to-26th keeps earlier idx
          float v = d; int idx = m;
#pragma unroll
          for (int s = 0; s < NBRC; ++s) csel_swap(v, idx, tv[s], ti[s]);
          thresh = tv[NBRC - 1];
        }
      }
    }
  }
#pragma unroll
  for (int s = 0; s < NBRC; ++s) nbr_cells[c * NBRC + s] = ti[s];
}

// ---------------------------------------------------------------------------
// Kernel D: per atom, stable top-8 over the 130 neighbor candidates, then the
// Vina-like energy terms; deterministic fixed-order block reduction.
// All coords + vdw staged in LDS (64 KB of the 320 KB/WGP) so the scattered
// candidate gathers hit LDS instead of global.
// ---------------------------------------------------------------------------
__global__ __launch_bounds__(128) void energy_kernel(
    const float* __restrict__ coords, const float* __restrict__ vdw,
    const float* __restrict__ weights, const int* __restrict__ cfa,
    const int* __restrict__ nbr_cells, const int* __restrict__ atoms_in_cells,
    float* __restrict__ partial) {
  __shared__ float s_x[NATOMS];
  __shared__ float s_y[NATOMS];
  __shared__ float s_z[NATOMS];
  __shared__ float s_w[NATOMS];
  __shared__ float s_red[128];

  for (int k = threadIdx.x; k < NATOMS; k += 128) {
    s_x[k] = coords[3 * k + 0];
    s_y[k] = coords[3 * k + 1];
    s_z[k] = coords[3 * k + 2];
    s_w[k] = vdw[k];
  }
  __syncthreads();

  const int i = blockIdx.x * blockDim.x + threadIdx.x;
  float fsum = 0.0f;
  if (i < NATOMS) {
    const float xi = s_x[i];
    const float yi = s_y[i];
    const float zi = s_z[i];
    const int cell = cfa[i];

    float tv[MNB]; int tj[MNB];
#pragma unroll
    for (int s = 0; s < MNB; ++s) { tv[s] = -__builtin_inff(); tj[s] = 0; }

    for (int nb = 0; nb < NBRC; ++nb) {
      const int nc = nbr_cells[cell * NBRC + nb];
#pragma unroll
      for (int s = 0; s < KPC; ++s) {
        const int j = atoms_in_cells[nc * KPC + s];
        const float dx = xi - s_x[j];
        const float dy = yi - s_y[j];
        const float dz = zi - s_z[j];
        float v = dx * dx + dy * dy + dz * dz;
        int jj = j;
#pragma unroll
        for (int t = 0; t < MNB; ++t) csel_swap(v, jj, tv[t], tj[t]);
      }
    }

    const float w0 = weights[0], w1 = weights[1], w2 = weights[2];
    const float w3 = weights[3], w4 = weights[4];
    const float vi = s_w[i];
#pragma unroll
    for (int t = 0; t < MNB; ++t) {
      const int j = tj[t];
      const float dx = xi - s_x[j];
      const float dy = yi - s_y[j];
      const float dz = zi - s_z[j];
      const float r = sqrtf(dx * dx + dy * dy + dz * dz + 1e-12f);
      const float d = r - vi - s_w[j];
      const float q1 = d * 2.0f;                 // d / 0.5
      const float g1 = __expf(-q1 * q1);
      const float q2 = (d - 3.0f) * 0.5f;        // (d - 3) / 2
      const float g2 = __expf(-q2 * q2);
      const float rep = (d < 0.0f) ? d * d : 0.0f;
      const float hyd = (d < 0.5f) ? 1.0f : ((d < 1.5f) ? 1.5f - d : 0.0f);
      const float hb  = (d < -0.7f) ? 1.0f : ((d < 0.0f) ? -d / 0.7f : 0.0f);
      const float hval = w0 * g1 + w1 * g2 + w2 * rep + w3 * hyd + w4 * hb;
      fsum += (d < 8.0f) ? hval : 0.0f;
    }
  }

  s_red[threadIdx.x] = fsum;
  __syncthreads();
#pragma unroll
  for (int off = 64; off > 0; off >>= 1) {
    if (threadIdx.x < off) s_red[threadIdx.x] += s_red[threadIdx.x + off];
    __syncthreads();
  }
  if (threadIdx.x == 0) partial[blockIdx.x] = s_red[0];
}

__global__ void finalize_kernel(const float* __restrict__ partial,
                                const float* __restrict__ weights,
                                const float* __restrict__ nrot,
                                float* __restrict__ out) {
  if (threadIdx.x == 0 && blockIdx.x == 0) {
    float c = 0.0f;
    for (int k = 0; k < NATOMS / 128; ++k) c += partial[k];   // fixed order
    out[0] = c / (1.0f + weights[5] * nrot[0]);
  }
}

// ---------------------------------------------------------------------------
// Launch
// ---------------------------------------------------------------------------
extern "C" void kernel_launch(void* const* d_in, const int* in_sizes, int n_in,
                              void* d_out, int out_size, void* d_ws, size_t ws_size,
                              hipStream_t stream) {
  const float* coords  = (const float*)d_in[0];  // (4096,3)
  const float* vdw     = (const float*)d_in[1];  // (4096,)
  const float* weights = (const float*)d_in[2];  // (6,)
  const float* nrot    = (const float*)d_in[3];  // (1,)
  float* out = (float*)d_out;                    // scalar

  // Workspace layout (~1 MB)
  int* aic  = (int*)d_ws;                   // NCELLS*KPC  atoms_in_cells
  int* cfa  = aic + NCELLS * KPC;           // NATOMS      cells_for_atoms
  int* nbrc = cfa + NATOMS;                 // NCELLS*NBRC neighbor cells
  float* partial = (float*)(nbrc + NCELLS * NBRC);  // NATOMS/128 block sums

  topk_atoms_per_cell_kernel<<<NCELLS / 16, 128, 0, stream>>>(coords, aic);
  cells_for_atoms_kernel<<<NATOMS / 128, 128, 0, stream>>>(coords, cfa);
  nbr_cells_kernel<<<(NCELLS + 127) / 128, 128, 0, stream>>>(nbrc);
  energy_kernel<<<NATOMS / 128, 128, 0, stream>>>(coords, vdw, weights, cfa,
                                                  nbrc, aic, partial);
  finalize_kernel<<<1, 32, 0, stream>>>(partial, weights, nrot, out);
}